// SmallLesionFocusedLoss_76416058131040
// MI455X (gfx1250) — compile-verified
//
#include <hip/hip_runtime.h>
#include <hip/hip_bf16.h>
#include <math.h>

// ---------------------------------------------------------------------------
// SmallLesionFocusedLoss for gfx1250 (MI455X, wave32).
//
// Roofline: ~290MB total traffic -> ~13us @ 23.3TB/s; FLOPs negligible.
// All dense reductions run on the matrix pipe (v_wmma_f32_16x16x4_f32 with an
// all-ones B operand: D[m][n] += sum_k A[m][k]); transcendentals use the raw
// hardware ops (v_exp_f32 / v_log_f32 / v_rcp_f32) since every argument is in
// the fixup-free range (log input in [1,2]; exp saturation is the desired
// sigmoid/BCE asymptote). Per-block LDS staging cuts same-address atomics 8x.
//
//   K0: zero 8 scalar accumulators in ws
//   K1: fused streaming pass (b128 loads, 4 px/lane/iter): sigmoid/BCE/focal;
//       5 WMMA reduction channels; init union-find labels + zero areas (b128)
//   K2: lock-free union-find merge (8-connectivity, half-neighborhood)
//   K3: path-compression flatten
//   K4: per-component area via atomicAdd on root slot
//   K5: streaming pass 2: WMMA-accumulated BCE over small-component pixels
//       (values zeroed when not small => EXEC all-ones at the WMMA)
//   K6: single-thread final combine -> scalar loss
// ---------------------------------------------------------------------------

#define ALPHA_C  0.7f
#define BETA_C   0.3f
#define SLW_C    10.0f
#define SMOOTH_C 1e-05f
#define AREA_TH  100u

#define LOG2E_C  1.4426950408889634f
#define LN2_C    0.6931471805599453f

typedef __attribute__((ext_vector_type(2))) float v2f;
typedef __attribute__((ext_vector_type(8))) float v8f;

// Raw hardware transcendentals (no libm range-fixup code).
__device__ __forceinline__ float hw_exp(float x) {           // e^x
    return __builtin_amdgcn_exp2f(x * LOG2E_C);              // v_exp_f32
}
__device__ __forceinline__ float hw_log1p_safe(float y) {    // ln(1+y), y in [0,1]
    return LN2_C * __builtin_amdgcn_logf(1.0f + y);          // v_log_f32
}
__device__ __forceinline__ float hw_rcp(float x) {
    return __builtin_amdgcn_rcpf(x);                         // v_rcp_f32
}

// D = A(16x4) * B(4x16) + C ; with B == all-ones every column of D carries the
// same running total of all 64 A elements fed per issue.
__device__ __forceinline__ v8f wmma_sum(v2f a, v2f ones, v8f c) {
    return __builtin_amdgcn_wmma_f32_16x16x4_f32(
        /*neg_a=*/false, a, /*neg_b=*/false, ones,
        /*c_mod=*/(short)0, c, /*reuse_a=*/false, /*reuse_b=*/false);
}

// Collapse a 16x16 f32 accumulator whose columns all hold the same total:
// per-lane sum of 8 VGPRs = column-half sum; add opposite 16-lane half.
__device__ __forceinline__ float wave_total(v8f a) {
    float s = 0.0f;
#pragma unroll
    for (int k = 0; k < 8; ++k) s += a[k];
    s += __shfl_xor(s, 16, 32);
    return s;
}

__device__ __forceinline__ float bce_fast(float x, float t) {
    // max(x,0) - x*t + log1p(exp(-|x|)); exp(-|x|) in (0,1] -> log arg in (1,2]
    return fmaxf(x, 0.0f) - x * t + hw_log1p_safe(hw_exp(-fabsf(x)));
}

__device__ __forceinline__ void px_terms(float x, float t,
                                         float& pt_prod, float& p_out,
                                         float& focal, float& bce) {
    const float p  = hw_rcp(1.0f + hw_exp(-x));          // sigmoid (rcp(inf)=0)
    const float b  = bce_fast(x, t);
    const bool  pos = (t == 1.0f);
    const float ptv = pos ? p : 1.0f - p;
    const float at  = pos ? ALPHA_C : 1.0f - ALPHA_C;
    float q = 1.0f - ptv;
    q = q * q; q = q * q;                                // (1-p_t)^4
    pt_prod = p * t;
    p_out   = p;
    focal   = at * q * b;
    bce     = b;
}

// ---------------------- K0: zero accumulators ------------------------------
__global__ void k_zero_acc(float* __restrict__ S) {
    if (threadIdx.x < 8) S[threadIdx.x] = 0.0f;
}

// ---------------------- K1: fused dense pass -------------------------------
// 4 consecutive pixels per lane per iteration (b128 loads). Grid sized so
// nquads % (gridDim*blockDim) == 0: no ragged tail, EXEC all-ones at WMMAs.
__global__ void k_main_pass(const float4* __restrict__ pred4,
                            const float4* __restrict__ targ4,
                            float* __restrict__ S,
                            int*   __restrict__ labels,
                            unsigned* __restrict__ areas,
                            int nquads) {
    __shared__ float red[8][5];
    const v2f ones = {1.0f, 1.0f};
    v8f accI = {}, accP = {}, accT = {}, accF = {}, accB = {};

    const int stride = gridDim.x * blockDim.x;
    for (int g = blockIdx.x * blockDim.x + threadIdx.x; g < nquads; g += stride) {
        __builtin_prefetch(pred4 + g + stride, 0, 0);   // global_prefetch_b8
        const float4 x = pred4[g];
        const float4 t = targ4[g];
        const int base = 4 * g;

        // union-find init + area zero (re-done every call: deterministic)
        int4 lab;
        lab.x = (t.x > 0.5f) ? base + 0 : -1;
        lab.y = (t.y > 0.5f) ? base + 1 : -1;
        lab.z = (t.z > 0.5f) ? base + 2 : -1;
        lab.w = (t.w > 0.5f) ? base + 3 : -1;
        ((int4*)labels)[g] = lab;
        ((uint4*)areas)[g] = make_uint4(0u, 0u, 0u, 0u);

        float i0, i1, i2, i3, p0, p1, p2, p3, f0, f1, f2, f3, b0, b1, b2, b3;
        px_terms(x.x, t.x, i0, p0, f0, b0);
        px_terms(x.y, t.y, i1, p1, f1, b1);
        px_terms(x.z, t.z, i2, p2, f2, b2);
        px_terms(x.w, t.w, i3, p3, f3, b3);

        v2f aIlo = {i0, i1}, aIhi = {i2, i3};
        v2f aPlo = {p0, p1}, aPhi = {p2, p3};
        v2f aTlo = {t.x, t.y}, aThi = {t.z, t.w};
        v2f aFlo = {f0, f1}, aFhi = {f2, f3};
        v2f aBlo = {b0, b1}, aBhi = {b2, b3};

        accI = wmma_sum(aIlo, ones, accI); accI = wmma_sum(aIhi, ones, accI);
        accP = wmma_sum(aPlo, ones, accP); accP = wmma_sum(aPhi, ones, accP);
        accT = wmma_sum(aTlo, ones, accT); accT = wmma_sum(aThi, ones, accT);
        accF = wmma_sum(aFlo, ones, accF); accF = wmma_sum(aFhi, ones, accF);
        accB = wmma_sum(aBlo, ones, accB); accB = wmma_sum(aBhi, ones, accB);
    }

    float w[5];
    w[0] = wave_total(accI);
    w[1] = wave_total(accP);
    w[2] = wave_total(accT);
    w[3] = wave_total(accF);
    w[4] = wave_total(accB);

    const int wid = threadIdx.x >> 5;
    if ((threadIdx.x & 31) == 0) {
#pragma unroll
        for (int c = 0; c < 5; ++c) red[wid][c] = w[c];
    }
    __syncthreads();
    if (threadIdx.x < 5) {            // one atomic per block per channel
        float s = 0.0f;
#pragma unroll
        for (int k = 0; k < 8; ++k) s += red[k][threadIdx.x];
        atomicAdd(&S[threadIdx.x], s);
    }
}

// ---------------------- union-find helpers ---------------------------------
__device__ __forceinline__ int uf_load(const int* labels, int i) {
    return __hip_atomic_load(&labels[i], __ATOMIC_RELAXED, __HIP_MEMORY_SCOPE_AGENT);
}

__device__ __forceinline__ int uf_find(const int* labels, int l) {
    int p = uf_load(labels, l);
    while (p != l) { l = p; p = uf_load(labels, l); }
    return l;
}

__device__ __forceinline__ void uf_union(int* labels, int a, int b) {
    while (true) {
        a = uf_find(labels, a);
        b = uf_find(labels, b);
        if (a == b) return;
        const int mn = (a < b) ? a : b;
        const int mx = (a < b) ? b : a;
        const int old = atomicMin(&labels[mx], mn);
        if (old == mx) return;   // successfully linked
        a = old; b = mn;         // lost race: retry with observed value
    }
}

// ---------------------- K2: merge (8-connectivity) -------------------------
__global__ void k_ccl_merge(int* __restrict__ labels, int n, int HW, int W) {
    const int i = blockIdx.x * blockDim.x + threadIdx.x;
    if (i >= n) return;
    if (uf_load(labels, i) < 0) return;
    const int r = (i % HW) / W;
    const int c = i % W;
    if (r > 0) {
        const int up = i - W;
        if (c > 0     && uf_load(labels, up - 1) >= 0) uf_union(labels, i, up - 1);
        if (             uf_load(labels, up)     >= 0) uf_union(labels, i, up);
        if (c < W - 1 && uf_load(labels, up + 1) >= 0) uf_union(labels, i, up + 1);
    }
    if (c > 0 && uf_load(labels, i - 1) >= 0) uf_union(labels, i, i - 1);
}

// ---------------------- K3: flatten ----------------------------------------
__global__ void k_ccl_compress(int* __restrict__ labels, int n) {
    const int i = blockIdx.x * blockDim.x + threadIdx.x;
    if (i >= n) return;
    const int l = labels[i];
    if (l < 0) return;
    labels[i] = uf_find(labels, l);
}

// ---------------------- K4: component areas --------------------------------
__global__ void k_ccl_area(const int* __restrict__ labels,
                           unsigned* __restrict__ areas, int n) {
    const int i = blockIdx.x * blockDim.x + threadIdx.x;
    if (i >= n) return;
    const int l = labels[i];
    if (l >= 0) atomicAdd(&areas[l], 1u);
}

// ---------------------- K5: small-lesion extra BCE sum ---------------------
__global__ void k_small_pass(const float4* __restrict__ pred4,
                             const float4* __restrict__ targ4,
                             const int* __restrict__ labels,
                             const unsigned* __restrict__ areas,
                             float* __restrict__ S, int nquads) {
    __shared__ float red[8];
    const v2f ones = {1.0f, 1.0f};
    v8f acc = {};
    const int stride = gridDim.x * blockDim.x;
    for (int g = blockIdx.x * blockDim.x + threadIdx.x; g < nquads; g += stride) {
        const int4  l = ((const int4*)labels)[g];
        const float4 x = pred4[g];
        const float4 t = targ4[g];

        const bool s0 = (l.x >= 0) && (areas[l.x] < AREA_TH);
        const bool s1 = (l.y >= 0) && (areas[l.y] < AREA_TH);
        const bool s2 = (l.z >= 0) && (areas[l.z] < AREA_TH);
        const bool s3 = (l.w >= 0) && (areas[l.w] < AREA_TH);

        const float b0 = bce_fast(x.x, t.x);
        const float b1 = bce_fast(x.y, t.y);
        const float b2 = bce_fast(x.z, t.z);
        const float b3 = bce_fast(x.w, t.w);

        v2f lo = {s0 ? b0 : 0.0f, s1 ? b1 : 0.0f};   // zeros keep EXEC uniform
        v2f hi = {s2 ? b2 : 0.0f, s3 ? b3 : 0.0f};
        acc = wmma_sum(lo, ones, acc);
        acc = wmma_sum(hi, ones, acc);
    }
    const float w = wave_total(acc);
    const int wid = threadIdx.x >> 5;
    if ((threadIdx.x & 31) == 0) red[wid] = w;
    __syncthreads();
    if (threadIdx.x == 0) {
        float s = 0.0f;
#pragma unroll
        for (int k = 0; k < 8; ++k) s += red[k];
        atomicAdd(&S[5], s);
    }
}

// ---------------------- K6: final combine ----------------------------------
__global__ void k_finalize(const float* __restrict__ S,
                           float* __restrict__ out, float invN) {
    if (blockIdx.x == 0 && threadIdx.x == 0) {
        const float inter = S[0], sp = S[1], st = S[2];
        const float fsum = S[3], bsum = S[4], extra = S[5];
        const float dice = 1.0f - (2.0f * inter + SMOOTH_C) / (sp + st + SMOOTH_C);
        const float FP = sp - inter;
        const float FN = st - inter;
        const float tversky = 1.0f - (inter + SMOOTH_C) /
                              (inter + ALPHA_C * FP + BETA_C * FN + SMOOTH_C);
        const float focal = fsum * invN;
        const float small = (bsum + (SLW_C - 1.0f) * extra) * invN;
        out[0] = 0.25f * (dice + focal + tversky + small);
    }
}

// ---------------------------------------------------------------------------
extern "C" void kernel_launch(void* const* d_in, const int* in_sizes, int n_in,
                              void* d_out, int out_size, void* d_ws, size_t ws_size,
                              hipStream_t stream) {
    const float* pred = (const float*)d_in[0];
    const float* targ = (const float*)d_in[1];
    float* out = (float*)d_out;

    const int n  = in_sizes[0];          // 8*1*1024*1024 = 8388608
    const int W  = 1024;
    const int HW = 1024 * 1024;
    const int nquads = n / 4;            // 2097152

    char* ws = (char*)d_ws;
    float*    S      = (float*)ws;                            // 8 floats
    int*      labels = (int*)(ws + 256);                      // n * 4 bytes
    unsigned* areas  = (unsigned*)(ws + 256 + (size_t)n * 4); // n * 4 bytes

    // Dense passes: 2048 blocks * 256 thr = 524288 threads; nquads = 2097152
    // -> exactly 4 full iterations per thread (no ragged tail).
    const int DB = 256, DG = 2048;
    const int blocks = (n + 255) / 256;

    k_zero_acc    <<<1, 32, 0, stream>>>(S);
    k_main_pass   <<<DG, DB, 0, stream>>>((const float4*)pred, (const float4*)targ,
                                          S, labels, areas, nquads);
    k_ccl_merge   <<<blocks, 256, 0, stream>>>(labels, n, HW, W);
    k_ccl_compress<<<blocks, 256, 0, stream>>>(labels, n);
    k_ccl_area    <<<blocks, 256, 0, stream>>>(labels, areas, n);
    k_small_pass  <<<DG, DB, 0, stream>>>((const float4*)pred, (const float4*)targ,
                                          labels, areas, S, nquads);
    k_finalize    <<<1, 1, 0, stream>>>(S, out, 1.0f / (float)n);
}